// descriptorNN_40029095199299
// MI455X (gfx1250) — compile-verified
//
#include <hip/hip_runtime.h>
#include <hip/hip_bf16.h>
#include <math.h>

// ---------------------------------------------------------------------------
// DeepMD-style descriptor network for MI455X (gfx1250, wave32).
//
// Stage 1: per-action embedding (VALU f32) + contraction via WMMA:
//          C = G^T R  (M=10,N=4,K=64) as 16 chained v_wmma_f32_16x16x4_f32;
//          D = C[:2,:] @ C^T, Dval via rank-1 correction (C - g0 x R0).
// Stage 2: batched 20->32->32->1 MLPs via WMMA f32 16x16x4 (16 actions/tile),
//          X tiles staged global->LDS with global_load_async_to_lds_b32.
// Stage 3: deterministic log_softmax + value reduction (no float atomics).
// ---------------------------------------------------------------------------

#define N_ACT_C   100000
#define ATOM_C    64
#define N_TILES   6250          // N_ACT_C / 16, exact
#define NB_RED    256

typedef __attribute__((ext_vector_type(2))) float v2f;
typedef __attribute__((ext_vector_type(8))) float v8f;
typedef const float __attribute__((address_space(1)))* gcf;
typedef float __attribute__((address_space(1)))* gf;
typedef __attribute__((address_space(3))) float* lf;

__device__ __forceinline__ float wave_sum32(float v) {
#pragma unroll
    for (int off = 16; off > 0; off >>= 1) v += __shfl_xor(v, off, 32);
    return v;
}

__device__ __forceinline__ v8f wmma4(v2f a, v2f b, v8f c) {
    return __builtin_amdgcn_wmma_f32_16x16x4_f32(
        false, a, false, b, (short)0, c, false, false);
}

// ---------------------------------------------------------------------------
// Kernel 1: one wave per action. Embedding in VALU f32; the 64-atom
// contraction runs on the WMMA pipe: C = G^T R, K=64 -> 16 k-steps.
// grid = 12500 blocks x 256 threads (8 waves/block), exact.
// LDS/wave: G[64][17] + R[64*5+16] + C/aux[64]  (~5.8 KB), x8 waves ~48 KB.
// ---------------------------------------------------------------------------
__global__ __launch_bounds__(256) void k1_descr(
    const float* Smat_, const float* Rraw_, const float* Msk_,
    const float* We1_, const float* be1_, const float* We2_, const float* be2_,
    float* Dout_, float* Dvout_)
{
    gcf Smat = (gcf)Smat_; gcf Rraw = (gcf)Rraw_; gcf Msk = (gcf)Msk_;
    gcf We1 = (gcf)We1_;  gcf be1 = (gcf)be1_;
    gcf We2 = (gcf)We2_;  gcf be2 = (gcf)be2_;
    gf  Dout = (gf)Dout_; gf  Dvout = (gf)Dvout_;

    __shared__ float wl[152];            // 0:We1[30] 30:be1[10] 40:We2[100] 140:be2[10]
    __shared__ float Gl[8][ATOM_C * 17]; // G^T staging, stride 17 (bank spread)
    __shared__ float Rl[8][ATOM_C * 5 + 16];
    __shared__ float Cl[8][64];          // 0..39: C[10][4]; 40..49: g0; 50..53: R0

    const int tid = threadIdx.x;
    if      (tid < 30)  wl[tid] = We1[tid];
    else if (tid < 40)  wl[tid] = be1[tid - 30];
    else if (tid < 140) wl[tid] = We2[tid - 40];
    else if (tid < 150) wl[tid] = be2[tid - 140];

    const int lane = tid & 31;
    const int w    = tid >> 5;
    const int a    = blockIdx.x * 8 + w;    // < 100000 exactly
    float* G  = Gl[w];
    float* R5 = Rl[w];
    float* cb = Cl[w];
    __syncthreads();

#pragma unroll
    for (int t = 0; t < 2; ++t) {
        const int j = lane + 32 * t;                        // atom index
        const float s = Smat[(size_t)a * ATOM_C + j];
        gcf rp = Rraw + ((size_t)a * ATOM_C + j) * 4;
        float R[4] = { s * rp[0], s * rp[1], s * rp[2], s * rp[3] };
        gcf mp = Msk + ((size_t)a * ATOM_C + j) * 3;
        const float sg0 = s * mp[0], sg1 = s * mp[1], sg2 = s * mp[2];

        float h[10];
#pragma unroll
        for (int e = 0; e < 10; ++e) {
            float v = wl[30 + e] + sg0 * wl[e] + sg1 * wl[10 + e] + sg2 * wl[20 + e];
            h[e] = v > 0.f ? v : 0.f;
        }
        float g[10];
#pragma unroll
        for (int e = 0; e < 10; ++e) {
            float acc = wl[140 + e];
#pragma unroll
            for (int f = 0; f < 10; ++f) acc += h[f] * wl[40 + f * 10 + e];
            g[e] = acc;
        }
#pragma unroll
        for (int e = 0; e < 10; ++e) G[j * 17 + e] = g[e];
#pragma unroll
        for (int c = 0; c < 4; ++c)  R5[j * 5 + c] = R[c];
        if (t == 0 && lane == 0) {       // atom 0 rank-1 term (excluded in Dval)
#pragma unroll
            for (int e = 0; e < 10; ++e) cb[40 + e] = g[e];
#pragma unroll
            for (int c = 0; c < 4; ++c)  cb[50 + c] = R[c];
        }
    }
    __syncthreads();

    // ---- C = G^T @ R : 16 chained WMMA k-steps over the 64 atoms ----
    const int mrow = lane & 15;           // emb index (rows of C)
    const int kb   = (lane >> 4) << 1;    // lanes 16-31 supply K+2,K+3
    const int ncol = lane & 15;           // coord index (cols of C, <4 valid)
    v8f c = {};
#pragma unroll
    for (int ks = 0; ks < 16; ++ks) {
        const int k0 = ks * 4 + kb;                       // <= 63
        v2f av; av[0] = G[k0 * 17 + mrow];       av[1] = G[(k0 + 1) * 17 + mrow];
        v2f bv; bv[0] = R5[k0 * 5 + ncol];       bv[1] = R5[(k0 + 1) * 5 + ncol];
        c = wmma4(av, bv, c);
    }

    // scatter the valid 10x4 block of C (C/D layout: M=r+8*(lane>=16), N=lane&15)
    if (lane < 4) {
#pragma unroll
        for (int r = 0; r < 8; ++r) cb[r * 4 + lane] = c[r];   // rows 0..7
    }
    if (lane >= 16 && lane < 20) {
        cb[32 + (lane - 16)] = c[0];                           // row 8
        cb[36 + (lane - 16)] = c[1];                           // row 9
    }
    __syncthreads();

    if (lane < 20) {
        const int k = lane / 10;
        const int n = lane - k * 10;
        const float g0k = cb[40 + k], g0n = cb[40 + n];
        float d = 0.f, dv = 0.f;
#pragma unroll
        for (int l = 0; l < 4; ++l) {
            const float Ckl = cb[k * 4 + l];
            const float Cnl = cb[n * 4 + l];
            const float r0  = cb[50 + l];
            d  += Ckl * Cnl;                                   // D = C[:2] @ C^T
            dv += (Ckl - g0k * r0) * (Cnl - g0n * r0);         // rank-1 corrected
        }
        Dout [(size_t)a * 20 + lane] = d;
        Dvout[(size_t)a * 20 + lane] = dv;
    }
}

// ---------------------------------------------------------------------------
// Kernel 2: WMMA-batched 20->32->32->1 MLP, 16 actions per wave-tile.
// X tiles staged global->LDS via async copy (ASYNCcnt).
// ---------------------------------------------------------------------------
__device__ float mlp16_wmma(const float* __restrict__ Dg,
                            gcf W1, gcf B1, gcf W2, gcf B2, gcf W3, gcf B3,
                            float* Xl, float* Hl, int tile, int lane)
{
    // ---- async-stage X tile [16][20] into LDS (stride 33) ----
    const unsigned long long sbase =
        (unsigned long long)(Dg + (size_t)tile * 320);
#pragma unroll
    for (int it = 0; it < 10; ++it) {
        const int idx = it * 32 + lane;            // < 320, contiguous in global
        const int m = idx / 20, cc = idx - m * 20;
        const unsigned ldsoff = (unsigned)(unsigned long long)(lf)&Xl[m * 33 + cc];
        const unsigned voff   = (unsigned)(idx * 4);
        asm volatile("global_load_async_to_lds_b32 %0, %1, %2"
                     :: "v"(ldsoff), "v"(voff), "s"(sbase) : "memory");
    }
    asm volatile("s_wait_asynccnt 0x0" ::: "memory");
    __syncthreads();

    const int mrow = lane & 15;
    const int kb   = (lane >> 4) << 1;
    const int ncol = lane & 15;
    const int hi   = (lane >> 4) * 8;

    // ---- layer 1: [16,20] @ [20,32], K = 5 steps of 4 ----
    v8f c0 = {}, c1 = {};
#pragma unroll
    for (int kk = 0; kk < 5; ++kk) {
        const int k0 = kk * 4 + kb;                // <= 19
        v2f av;  av[0] = Xl[mrow * 33 + k0]; av[1] = Xl[mrow * 33 + k0 + 1];
        v2f b0;  b0[0] = W1[k0 * 32 + ncol];       b0[1] = W1[(k0 + 1) * 32 + ncol];
        v2f bh;  bh[0] = W1[k0 * 32 + ncol + 16];  bh[1] = W1[(k0 + 1) * 32 + ncol + 16];
        c0 = wmma4(av, b0, c0);
        c1 = wmma4(av, bh, c1);
    }
    __syncthreads();
#pragma unroll
    for (int r = 0; r < 8; ++r) {
        float v0 = c0[r] + B1[ncol];
        float v1 = c1[r] + B1[ncol + 16];
        Hl[(r + hi) * 33 + ncol]      = v0 > 0.f ? v0 : 0.f;
        Hl[(r + hi) * 33 + ncol + 16] = v1 > 0.f ? v1 : 0.f;
    }
    __syncthreads();

    // ---- layer 2: [16,32] @ [32,32], K = 8 steps of 4 ----
    c0 = (v8f){}; c1 = (v8f){};
#pragma unroll
    for (int kk = 0; kk < 8; ++kk) {
        const int k0 = kk * 4 + kb;                // <= 31
        v2f av;  av[0] = Hl[mrow * 33 + k0]; av[1] = Hl[mrow * 33 + k0 + 1];
        v2f b0;  b0[0] = W2[k0 * 32 + ncol];       b0[1] = W2[(k0 + 1) * 32 + ncol];
        v2f bh;  bh[0] = W2[k0 * 32 + ncol + 16];  bh[1] = W2[(k0 + 1) * 32 + ncol + 16];
        c0 = wmma4(av, b0, c0);
        c1 = wmma4(av, bh, c1);
    }
    __syncthreads();                               // X region free; reuse for H2
#pragma unroll
    for (int r = 0; r < 8; ++r) {
        float v0 = c0[r] + B2[ncol];
        float v1 = c1[r] + B2[ncol + 16];
        Xl[(r + hi) * 33 + ncol]      = v0 > 0.f ? v0 : 0.f;
        Xl[(r + hi) * 33 + ncol + 16] = v1 > 0.f ? v1 : 0.f;
    }
    __syncthreads();

    // ---- layer 3: [16,32] @ [32,1] ----
    float res = 0.f;
    if (lane < 16) {
        res = B3[0];
#pragma unroll
        for (int n = 0; n < 32; ++n) res += Xl[lane * 33 + n] * W3[n];
    }
    __syncthreads();
    return res;
}

__global__ __launch_bounds__(64) void k2_mlp(
    const float* __restrict__ Dg,  const float* __restrict__ Dvg,
    const float* Wf1_, const float* bf1_, const float* Wf2_, const float* bf2_,
    const float* Wf3_, const float* bf3_,
    const float* Wv1_, const float* bv1_, const float* Wv2_, const float* bv2_,
    const float* Wv3_, const float* bv3_,
    float* __restrict__ logits, float* __restrict__ vpart)
{
    __shared__ float smem[2][2][16 * 33];
    const int lane = threadIdx.x & 31;
    const int w    = threadIdx.x >> 5;
    const int tile = blockIdx.x * 2 + w;    // grid 3125 x 2 waves = 6250 exact

    const float pl = mlp16_wmma(Dg, (gcf)Wf1_, (gcf)bf1_, (gcf)Wf2_, (gcf)bf2_,
                                (gcf)Wf3_, (gcf)bf3_, smem[w][0], smem[w][1], tile, lane);
    if (lane < 16) logits[tile * 16 + lane] = pl;

    const float vv = mlp16_wmma(Dvg, (gcf)Wv1_, (gcf)bv1_, (gcf)Wv2_, (gcf)bv2_,
                                (gcf)Wv3_, (gcf)bv3_, smem[w][0], smem[w][1], tile, lane);
    float val = (lane < 16) ? vv : 0.f;
    val = wave_sum32(val);
    if (lane == 0) vpart[tile] = val;
}

// ---------------------------------------------------------------------------
// Deterministic reductions for log_softmax + value.
// ---------------------------------------------------------------------------
__global__ __launch_bounds__(256) void k3_max(const float* __restrict__ logits,
                                              float* __restrict__ pmax)
{
    __shared__ float s[256];
    const int t = threadIdx.x;
    float m = -INFINITY;
    for (int i = blockIdx.x * 256 + t; i < N_ACT_C; i += NB_RED * 256)
        m = fmaxf(m, logits[i]);
    s[t] = m; __syncthreads();
    for (int off = 128; off > 0; off >>= 1) {
        if (t < off) s[t] = fmaxf(s[t], s[t + off]);
        __syncthreads();
    }
    if (t == 0) pmax[blockIdx.x] = s[0];
}

__global__ __launch_bounds__(256) void k4_sumexp(const float* __restrict__ logits,
                                                 const float* __restrict__ pmax,
                                                 float* __restrict__ psum)
{
    __shared__ float s[256];
    const int t = threadIdx.x;
    float gm = -INFINITY;
    for (int j = 0; j < NB_RED; ++j) gm = fmaxf(gm, pmax[j]);
    float acc = 0.f;
    for (int i = blockIdx.x * 256 + t; i < N_ACT_C; i += NB_RED * 256)
        acc += expf(logits[i] - gm);
    s[t] = acc; __syncthreads();
    for (int off = 128; off > 0; off >>= 1) {
        if (t < off) s[t] += s[t + off];
        __syncthreads();
    }
    if (t == 0) psum[blockIdx.x] = s[0];
}

__global__ __launch_bounds__(256) void k5_final(const float* __restrict__ logits,
                                                const float* __restrict__ pmax,
                                                const float* __restrict__ psum,
                                                const float* __restrict__ vpart,
                                                float* __restrict__ out)
{
    __shared__ float s[256];
    const int t = threadIdx.x;
    float gm = -INFINITY, tot = 0.f;
    for (int j = 0; j < NB_RED; ++j) gm = fmaxf(gm, pmax[j]);
    for (int j = 0; j < NB_RED; ++j) tot += psum[j];
    const float lse = gm + logf(tot);

    const int i = blockIdx.x * 256 + t;
    if (i < N_ACT_C) out[i] = logits[i] - lse;

    if (blockIdx.x == 0) {                       // value = fixed-order sum
        float v = 0.f;
        for (int j = t; j < N_TILES; j += 256) v += vpart[j];
        s[t] = v; __syncthreads();
        for (int off = 128; off > 0; off >>= 1) {
            if (t < off) s[t] += s[t + off];
            __syncthreads();
        }
        if (t == 0) out[N_ACT_C] = s[0];
    }
}

// ---------------------------------------------------------------------------
extern "C" void kernel_launch(void* const* d_in, const int* in_sizes, int n_in,
                              void* d_out, int out_size, void* d_ws, size_t ws_size,
                              hipStream_t stream) {
    (void)in_sizes; (void)n_in; (void)out_size; (void)ws_size;
    const float* Smat = (const float*)d_in[0];
    const float* Rraw = (const float*)d_in[1];
    const float* Msk  = (const float*)d_in[2];
    float* out = (float*)d_out;

    float* ws     = (float*)d_ws;                 // needs ~16.5 MB
    float* Dws    = ws;                           // [100000,20]
    float* Dvws   = ws + 2000000;                 // [100000,20]
    float* logits = ws + 4000000;                 // [100000]
    float* vpart  = ws + 4100000;                 // [6250]
    float* pmax   = ws + 4106250;                 // [256]
    float* psum   = pmax + NB_RED;                // [256]

    k1_descr<<<N_ACT_C / 8, 256, 0, stream>>>(
        Smat, Rraw, Msk,
        (const float*)d_in[3], (const float*)d_in[4],
        (const float*)d_in[5], (const float*)d_in[6], Dws, Dvws);

    k2_mlp<<<N_TILES / 2, 64, 0, stream>>>(
        Dws, Dvws,
        (const float*)d_in[7],  (const float*)d_in[8],
        (const float*)d_in[9],  (const float*)d_in[10],
        (const float*)d_in[11], (const float*)d_in[12],
        (const float*)d_in[13], (const float*)d_in[14],
        (const float*)d_in[15], (const float*)d_in[16],
        (const float*)d_in[17], (const float*)d_in[18],
        logits, vpart);

    k3_max   <<<NB_RED, 256, 0, stream>>>(logits, pmax);
    k4_sumexp<<<NB_RED, 256, 0, stream>>>(logits, pmax, psum);
    k5_final <<<(N_ACT_C + 255) / 256, 256, 0, stream>>>(logits, pmax, psum, vpart, out);
}